// NUFFT_3917010174180
// MI455X (gfx1250) — compile-verified
//
#include <hip/hip_runtime.h>
#include <math.h>

// ---------------------------------------------------------------------------
// NUFFT forward for MI455X (gfx1250, wave32).
// Spectral transform = DFT-by-GEMM with V_WMMA_F32_16X16X4_F32.
// B-operand (symmetric DFT matrix) staged to LDS per K-chunk with
// GLOBAL_LOAD_ASYNC_TO_LDS_B64 (ASYNCcnt) -- CDNA5 async data mover path.
// ---------------------------------------------------------------------------

typedef float v2f __attribute__((ext_vector_type(2)));
typedef float v8f __attribute__((ext_vector_type(8)));

#define NDIM     320
#define GDIM     640
#define GGE      (GDIM * GDIM)          // 409600
#define NBATCH   2
#define NCOIL    12
#define NIMG     (NBATCH * NCOIL)       // 24
#define KPTS     (400 * 640)            // 256000
#define JW       6
#define ALPHA_KB 14.04f                 // 2.34 * J
#define NSHIFT_F 160.0f                 // N // 2
#define PI_F     3.14159265358979323846f

#define KC       160                    // K-chunk staged in LDS
#define CHUNKS   (GDIM / KC)            // 4
#define PLANEF   (16 * KC)              // floats per LDS plane (2560)
#define LDS_BYTES (2 * PLANEF * 4)      // 20480 B dynamic LDS

// ---------------- device math helpers --------------------------------------

__device__ __forceinline__ float i0f(float x) {
    x = fabsf(x);
    if (x < 3.75f) {
        float t = x / 3.75f; t = t * t;
        return 1.0f + t * (3.5156229f + t * (3.0899424f + t * (1.2067492f
             + t * (0.2659732f + t * (0.0360768f + t * 0.0045813f)))));
    } else {
        float ti = 3.75f / x;
        return (expf(x) * rsqrtf(x)) * (0.39894228f
             + ti * (0.01328592f + ti * (0.00225319f + ti * (-0.00157565f
             + ti * (0.00916281f + ti * (-0.02057706f + ti * (0.02635537f
             + ti * (-0.01647633f + ti * 0.00392377f))))))));
    }
}

__device__ __forceinline__ float kb_weight(float d, float inv_i0a) {
    float u = (2.0f * d) / (float)JW;
    float f = fmaxf(1.0f - u * u, 0.0f);
    return i0f(ALPHA_KB * sqrtf(f)) * inv_i0a;
}

// apodization 1/FT(kb) along one axis (Fessler convention)
__device__ __forceinline__ float apod1(int i, float inv_i0a) {
    float n   = ((float)i - (float)(NDIM - 1) * 0.5f) / (float)GDIM;
    float pjn = PI_F * (float)JW * n;
    float t   = pjn * pjn - ALPHA_KB * ALPHA_KB;
    float s   = sqrtf(fabsf(t));
    float core = (t >= 0.0f) ? (sinf(s) / fmaxf(s, 1e-12f))
                             : (sinhf(s) / fmaxf(s, 1e-12f));
    float ft = core * (float)JW * inv_i0a;
    return 1.0f / ft;
}

// ---------------- kernel 1: DFT matrix (symmetric, ortho-normalized) -------

__global__ __launch_bounds__(256) void gen_dftmat(float* __restrict__ fRe,
                                                  float* __restrict__ fIm) {
    int tid = blockIdx.x * 256 + threadIdx.x;       // GGE threads exactly
    int n = tid / GDIM;
    int k = tid - n * GDIM;
    int p = (int)(((long long)n * (long long)k) % GDIM);
    float ang = (-2.0f * PI_F / (float)GDIM) * (float)p;
    float s, c;
    sincosf(ang, &s, &c);
    float nrm = rsqrtf((float)GDIM);                // 1/sqrt(G) per 1-D pass
    fRe[tid] = c * nrm;
    fIm[tid] = s * nrm;
}

// ---------------- kernel 2: coil images * apodization, zero-padded ---------

__global__ __launch_bounds__(256) void prep_grid(const float* __restrict__ inp,
                                                 const float* __restrict__ smaps,
                                                 float* __restrict__ aRe,
                                                 float* __restrict__ aIm) {
    size_t tid = (size_t)blockIdx.x * 256 + threadIdx.x;   // NIMG*GGE threads
    int img = (int)(tid / GGE);
    int rem = (int)(tid - (size_t)img * GGE);
    int i = rem / GDIM;
    int j = rem - i * GDIM;
    float re = 0.0f, im = 0.0f;
    if (i < NDIM && j < NDIM) {
        int b = img / NCOIL;
        int c = img - b * NCOIL;
        size_t xoff = (((size_t)b * NDIM + i) * NDIM + j) * 2;
        size_t soff = ((((size_t)b * NCOIL + c) * NDIM + i) * NDIM + j) * 2;
        float xr = inp[xoff], xi = inp[xoff + 1];
        float sr = smaps[soff], si = smaps[soff + 1];
        float inv_i0a = 1.0f / i0f(ALPHA_KB);
        float sc = apod1(i, inv_i0a) * apod1(j, inv_i0a);
        re = (xr * sr - xi * si) * sc;
        im = (xr * si + xi * sr) * sc;
    }
    aRe[tid] = re;
    aIm[tid] = im;
}

// ---------------- kernel 3: one 640x640 complex GEMM pass, O = (A*F)^T -----
// Block = 8 waves sharing one tn (16 output cols); wave w owns tm = grp*8+w.
// The 16-row F strip for tn is staged into dynamic LDS per 160-wide K-chunk
// using GLOBAL_LOAD_ASYNC_TO_LDS_B64 (ASYNCcnt), then B fragments are served
// by ds_load_b64 while A fragments stream from global.
// Dynamic LDS is the only LDS in this kernel, so its base offset is 0 and the
// async-load LDS destination address is a plain integer.
__global__ __launch_bounds__(256) void dft_pass(const float* __restrict__ aRe,
                                                const float* __restrict__ aIm,
                                                const float* __restrict__ fRe,
                                                const float* __restrict__ fIm,
                                                float* __restrict__ oRe,
                                                float* __restrict__ oIm) {
    extern __shared__ float smem[];                  // [2][16][KC]

    const int lane = threadIdx.x & 31;
    const int wave = threadIdx.x >> 5;
    const int tn   = blockIdx.x % 40;                // shared by all 8 waves
    const int grp  = blockIdx.x / 40;                // 0..4
    const int tm   = grp * 8 + wave;                 // 0..39

    const int l15    = lane & 15;
    const int khalf  = (lane >> 4) * 2;              // ISA A-frag K layout
    const int arow   = tm * 16 + l15;
    const int tnRow0 = tn * 16;

    const float* aR = aRe + (size_t)arow * GDIM + khalf;
    const float* aI = aIm + (size_t)arow * GDIM + khalf;

    v8f cr = {};
    v8f ci = {};

    for (int ch = 0; ch < CHUNKS; ++ch) {
        const int c0 = ch * KC;

        // ---- async stage: F[tnRow0..+16][c0..c0+KC) both planes -> LDS ----
        // 2 planes * 16 rows * 80 float2 = 2560 b64 transfers, 10/thread.
        for (int idx = threadIdx.x; idx < 2 * 16 * (KC / 2); idx += 256) {
            const int plane = idx / (16 * (KC / 2));
            const int e     = idx - plane * (16 * (KC / 2));
            const int row   = e / (KC / 2);
            const int c2    = (e - row * (KC / 2)) * 2;
            const float* src = (plane ? fIm : fRe)
                             + (size_t)(tnRow0 + row) * GDIM + c0 + c2;
            unsigned lds_addr = (unsigned)(((plane * 16 + row) * KC + c2) * 4);
            unsigned long long ga = (unsigned long long)(uintptr_t)src;
            asm volatile("global_load_async_to_lds_b64 %0, %1, off"
                         :: "v"(lds_addr), "v"(ga) : "memory");
        }
        asm volatile("s_wait_asynccnt 0x0" ::: "memory");
        __syncthreads();

        // ---- compute this K-chunk: A from global, B from LDS --------------
        const float* aRc = aR + c0;
        const float* aIc = aI + c0;
        const float* bRl = smem + l15 * KC + khalf;            // plane 0 (re)
        const float* bIl = smem + PLANEF + l15 * KC + khalf;   // plane 1 (im)

        for (int k0 = 0; k0 < KC; k0 += 4) {
            v2f ar = *(const v2f*)(aRc + k0);
            v2f ai = *(const v2f*)(aIc + k0);
            v2f br = *(const v2f*)(bRl + k0);
            v2f bi = *(const v2f*)(bIl + k0);
            v2f nbi = -bi;                       // f32 WMMA has no A/B neg
            // Cr += Ar*Br - Ai*Bi ; Ci += Ar*Bi + Ai*Br
            cr = __builtin_amdgcn_wmma_f32_16x16x4_f32(false, ar, false, br,
                                                       (short)0, cr, false, false);
            cr = __builtin_amdgcn_wmma_f32_16x16x4_f32(false, ai, false, nbi,
                                                       (short)0, cr, false, false);
            ci = __builtin_amdgcn_wmma_f32_16x16x4_f32(false, ar, false, bi,
                                                       (short)0, ci, false, false);
            ci = __builtin_amdgcn_wmma_f32_16x16x4_f32(false, ai, false, br,
                                                       (short)0, ci, false, false);
        }
        __syncthreads();                          // protect LDS for next chunk
    }

    // C/D layout: VGPR v, lanes0-15 -> (M=v, N=lane); lanes16-31 -> (M=v+8).
    // Transposed store: element (M,N) -> O[N][M]; per lane 8 consecutive M.
    const int nrow = tn * 16 + l15;
    const int mcol = tm * 16 + ((lane >> 4) << 3);
    float4* pr = (float4*)(oRe + (size_t)nrow * GDIM + mcol);
    float4* pi = (float4*)(oIm + (size_t)nrow * GDIM + mcol);
    float4 t;
    t.x = cr[0]; t.y = cr[1]; t.z = cr[2]; t.w = cr[3]; pr[0] = t;
    t.x = cr[4]; t.y = cr[5]; t.z = cr[6]; t.w = cr[7]; pr[1] = t;
    t.x = ci[0]; t.y = ci[1]; t.z = ci[2]; t.w = ci[3]; pi[0] = t;
    t.x = ci[4]; t.y = ci[5]; t.z = ci[6]; t.w = ci[7]; pi[1] = t;
}

// ---------------- kernel 4: KB interpolation + phase + dcomp ---------------

__global__ __launch_bounds__(256) void interp(const float* __restrict__ ktraj,
                                              const float* __restrict__ dcomp,
                                              const float* __restrict__ gRe,
                                              const float* __restrict__ gIm,
                                              float* __restrict__ out) {
    int tid = blockIdx.x * 256 + threadIdx.x;
    if (tid >= NBATCH * KPTS) return;
    int b = tid / KPTS;
    int k = tid - b * KPTS;

    const float inv_i0a = 1.0f / i0f(ALPHA_KB);
    float kx = ktraj[((size_t)b * 2 + 0) * KPTS + k];
    float ky = ktraj[((size_t)b * 2 + 1) * KPTS + k];
    float tmx = kx * ((float)GDIM / (2.0f * PI_F));
    float tmy = ky * ((float)GDIM / (2.0f * PI_F));
    float koffx = floorf(tmx - (float)JW * 0.5f);
    float koffy = floorf(tmy - (float)JW * 0.5f);

    float wx[JW], wy[JW];
    int   ix[JW], iy[JW];
#pragma unroll
    for (int j = 0; j < JW; ++j) {
        float kpx = koffx + (float)(j + 1);
        float kpy = koffy + (float)(j + 1);
        wx[j] = kb_weight(tmx - kpx, inv_i0a);
        wy[j] = kb_weight(tmy - kpy, inv_i0a);
        int iix = (int)kpx, iiy = (int)kpy;          // integral-valued floats
        ix[j] = ((iix % GDIM) + GDIM) % GDIM;
        iy[j] = ((iiy % GDIM) + GDIM) % GDIM;
    }

    float accR[NCOIL], accI[NCOIL];
#pragma unroll
    for (int c = 0; c < NCOIL; ++c) { accR[c] = 0.0f; accI[c] = 0.0f; }

    const size_t imgbase = (size_t)b * NCOIL * GGE;
#pragma unroll
    for (int jx = 0; jx < JW; ++jx) {
        const size_t rowoff = (size_t)ix[jx] * GDIM;
#pragma unroll
        for (int jy = 0; jy < JW; ++jy) {
            const float wgt = wx[jx] * wy[jy];
            const size_t flat = rowoff + (size_t)iy[jy];
#pragma unroll
            for (int c = 0; c < NCOIL; ++c) {
                const size_t off = imgbase + (size_t)c * GGE + flat;
                accR[c] = fmaf(gRe[off], wgt, accR[c]);
                accI[c] = fmaf(gIm[off], wgt, accI[c]);
            }
        }
    }

    float ps, pc;
    sincosf(NSHIFT_F * (kx + ky), &ps, &pc);
    float sd = sqrtf(dcomp[(size_t)b * KPTS + k]);
#pragma unroll
    for (int c = 0; c < NCOIL; ++c) {
        float orr = (accR[c] * pc - accI[c] * ps) * sd;
        float oii = (accR[c] * ps + accI[c] * pc) * sd;
        size_t o = ((((size_t)b * NCOIL + c) * KPTS) + k) * 2;
        out[o]     = orr;
        out[o + 1] = oii;
    }
}

// ---------------- host-side launch sequence --------------------------------

extern "C" void kernel_launch(void* const* d_in, const int* in_sizes, int n_in,
                              void* d_out, int out_size, void* d_ws, size_t ws_size,
                              hipStream_t stream) {
    const float* inp   = (const float*)d_in[0];   // (B,1,N,N,2)
    const float* smaps = (const float*)d_in[1];   // (B,NC,N,N,2)
    const float* ktraj = (const float*)d_in[2];   // (B,2,K)
    const float* dcomp = (const float*)d_in[3];   // (B,1,K)
    float* out = (float*)d_out;                   // (B,NC,K,2)

    // Workspace layout (floats): F(2*GGE) | grids(2*24*GGE) | temp(2*GGE)
    float* w   = (float*)d_ws;
    float* fRe = w;
    float* fIm = fRe + (size_t)GGE;
    float* aRe = fIm + (size_t)GGE;
    float* aIm = aRe + (size_t)NIMG * GGE;
    float* tRe = aIm + (size_t)NIMG * GGE;
    float* tIm = tRe + (size_t)GGE;

    gen_dftmat<<<GGE / 256, 256, 0, stream>>>(fRe, fIm);
    prep_grid<<<(NIMG * GGE) / 256, 256, 0, stream>>>(inp, smaps, aRe, aIm);

    // X = F * x * F  via two right-multiplies with transposed stores:
    //   pass1: T = (x*F)^T ; pass2: X = (T*F)^T  (row-major X[k1][k2]).
    for (int img = 0; img < NIMG; ++img) {
        const size_t off = (size_t)img * GGE;
        dft_pass<<<200, 256, LDS_BYTES, stream>>>(aRe + off, aIm + off,
                                                  fRe, fIm, tRe, tIm);
        dft_pass<<<200, 256, LDS_BYTES, stream>>>(tRe, tIm, fRe, fIm,
                                                  aRe + off, aIm + off);
    }

    interp<<<(NBATCH * KPTS + 255) / 256, 256, 0, stream>>>(ktraj, dcomp,
                                                            aRe, aIm, out);
}